// Policy_1838246002729
// MI455X (gfx1250) — compile-verified
//
#include <hip/hip_runtime.h>
#include <math.h>

typedef __attribute__((ext_vector_type(2))) float v2f;
typedef __attribute__((ext_vector_type(8))) float v8f;

#define NNODES 200000
#define NTOT   200001
#define NTPAD  200016          // padded to multiple of 16 rows
#define HD     128
#define NEDGE  1600000
#define NACT   200000
#define EFD    514
#define LLEAF  126

// ---------------------------------------------------------------------------
// degree / normalization
// ---------------------------------------------------------------------------
__global__ void deg_init_k(float* deg) {
  int i = blockIdx.x * blockDim.x + threadIdx.x;
  if (i < NTPAD) deg[i] = 1.0f;      // self-loop
}

__global__ void deg_accum_k(const int* __restrict__ ei, float* __restrict__ deg) {
  int i = blockIdx.x * blockDim.x + threadIdx.x;
  if (i >= NEDGE) return;
  int s = ei[i];
  int d = ei[NEDGE + i];
  atomicAdd(&deg[s], 1.0f);          // symmetrized: each directed edge hits both
  atomicAdd(&deg[d], 1.0f);
}

__global__ void isd_k(float* degisd) {
  int i = blockIdx.x * blockDim.x + threadIdx.x;
  if (i < NTPAD) degisd[i] = rsqrtf(degisd[i]);
}

// ---------------------------------------------------------------------------
// GEMM: Out[rows x 128] = A[rows x 128] @ W[128 x 128], fp32 WMMA 16x16x4.
// One block per 16-row tile; 8 waves, each wave owns one 16-col tile.
// A tile staged in (dynamic) LDS via GLOBAL_LOAD_ASYNC_TO_LDS_B128 with a
// 132-float row stride (bank-conflict free). Dynamic LDS => byte offset 0
// base, so the raw per-lane offsets handed to the async op are exact.
// ---------------------------------------------------------------------------
__global__ __launch_bounds__(256)
void gemm128_k(const float* __restrict__ A, const float* __restrict__ W,
               float* __restrict__ Out, int rowTiles) {
  extern __shared__ float As[];      // 16*132 floats = 8448 bytes, LDS offset 0
  int tile = blockIdx.x;
  if (tile >= rowTiles) return;
  size_t rowBase = (size_t)tile * 16;

  // async-stage 16x128 A tile: 512 x 16B chunks, 2 per thread, ASYNCcnt path
  {
    const float* base = A + rowBase * HD;
    #pragma unroll
    for (int it = 0; it < 2; ++it) {
      int i = threadIdx.x + it * 256;          // chunk index 0..511
      int r = (i * 4) >> 7, c = (i * 4) & 127;
      unsigned ldsoff = (unsigned)((r * 132 + c) * 4);
      const float* gp = base + i * 4;
      asm volatile("global_load_async_to_lds_b128 %0, %1, off"
                   :: "v"(ldsoff), "v"(gp) : "memory");
    }
    asm volatile("s_wait_asynccnt 0x0" ::: "memory");
  }
  __syncthreads();

  int lane = threadIdx.x & 31;
  int wave = threadIdx.x >> 5;
  int n16  = lane & 15;
  int hi   = lane >> 4;        // 0: lanes 0-15 (K 0,1 / M 0-7), 1: lanes 16-31 (K 2,3 / M 8-15)
  int koff = hi * 2;
  int wn   = wave * 16;

  v8f acc = {0.f,0.f,0.f,0.f,0.f,0.f,0.f,0.f};
  const float* Am = &As[n16 * 132];
  #pragma unroll 4
  for (int k = 0; k < 128; k += 4) {
    v2f a, b;
    a.x = Am[k + koff];
    a.y = Am[k + koff + 1];
    const float* wp = W + (size_t)(k + koff) * HD + wn + n16;
    b.x = wp[0];
    b.y = wp[HD];
    acc = __builtin_amdgcn_wmma_f32_16x16x4_f32(false, a, false, b,
                                                (short)0, acc, false, false);
  }
  float* op = Out + (rowBase + (size_t)hi * 8) * HD + wn + n16;
  #pragma unroll
  for (int r = 0; r < 8; ++r) op[(size_t)r * HD] = acc[r];
}

// focal node row of m1: one-hot(cf) + bit at 127  =>  W[cf,:] + W[127,:]
__global__ void focal_k(const float* __restrict__ Wg1, const int* __restrict__ cfp,
                        float* __restrict__ m1) {
  int j = threadIdx.x;
  int cf = *cfp;
  m1[(size_t)(NTOT - 1) * HD + j] = Wg1[(size_t)cf * HD + j] + Wg1[(size_t)127 * HD + j];
}

// hb = m1 * isd^2 + bias  (rows >= NTOT zeroed for GEMM padding)
__global__ void combine1_k(const float* __restrict__ m1, const float* __restrict__ isd,
                           const float* __restrict__ b, float* __restrict__ hb) {
  size_t i = (size_t)blockIdx.x * blockDim.x + threadIdx.x;
  if (i >= (size_t)NTPAD * HD) return;
  int row = (int)(i >> 7);
  float v = 0.f;
  if (row < NTOT) { float s = isd[row]; v = m1[i] * s * s + b[i & 127]; }
  hb[i] = v;
}

// ne = m2 * isd^2 + bias   (only the NTOT real rows; lives in d_out)
__global__ void combine2_k(const float* __restrict__ m2, const float* __restrict__ isd,
                           const float* __restrict__ b, float* __restrict__ ne) {
  size_t i = (size_t)blockIdx.x * blockDim.x + threadIdx.x;
  if (i >= (size_t)NTOT * HD) return;
  int row = (int)(i >> 7);
  float s = isd[row];
  ne[i] = m2[i] * s * s + b[i & 127];
}

// scatter-add over both directions: out[d] += coef * m[s]
__global__ void scatter_k(const int* __restrict__ ei, const float* __restrict__ isd,
                          const float* __restrict__ m, float* __restrict__ out) {
  long long tid = (long long)blockIdx.x * blockDim.x + threadIdx.x;
  if (tid >= 2LL * NEDGE * 32) return;
  int c = (int)(tid & 31);
  long long eg = tid >> 5;
  int s, d;
  if (eg < NEDGE) { s = ei[eg];           d = ei[NEDGE + eg]; }
  else            { long long e = eg - NEDGE; s = ei[NEDGE + e]; d = ei[e]; }
  // prefetch the edge-index stream ahead (global_prefetch_b8)
  __builtin_prefetch(ei + (eg & (NEDGE - 1)) + 8192, 0, 1);
  float co = isd[s] * isd[d];
  const float4 v = *(const float4*)(m + (size_t)s * HD + c * 4);
  float* o = out + (size_t)d * HD + c * 4;
  atomicAdd(o + 0, v.x * co);
  atomicAdd(o + 1, v.y * co);
  atomicAdd(o + 2, v.z * co);
  atomicAdd(o + 3, v.w * co);
}

__global__ void relu_k(float* h) {
  size_t i = (size_t)blockIdx.x * blockDim.x + threadIdx.x;
  if (i < (size_t)NTPAD * HD) h[i] = fmaxf(h[i], 0.f);
}

// ---------------------------------------------------------------------------
// Fused head: build 16x514 edge-feature tile in LDS, emit it, then
// z1 = elu(ef@Wh1+b1), z2 = elu(z1@Wh2+b2), logit = z2@Wh3+b3, z1/z2 via WMMA, logit via LDS atomics.
// Block = 128 threads (4 waves), each wave handles 2 of the 8 column tiles.
// ---------------------------------------------------------------------------
__device__ __forceinline__ float elu1(float x) { return x > 0.f ? x : expm1f(x); }

__global__ __launch_bounds__(128)
void head_k(const float* __restrict__ ne, const int* __restrict__ bc,
            const float* __restrict__ tv, const float* __restrict__ isr,
            const float* __restrict__ Wh1, const float* __restrict__ bh1,
            const float* __restrict__ Wh2, const float* __restrict__ bh2,
            const float* __restrict__ Wh3, const float* __restrict__ bh3,
            float* __restrict__ o_ef, float* __restrict__ o_log) {
  __shared__ float hf[HD];
  __shared__ float ef[16 * 518];     // 512 WMMA columns (stride 518: conflict-free)
  __shared__ float z1[16 * 132];
  __shared__ float tnl[16], irl[16], lgs[16];
  __shared__ int   child[16];

  int tid = threadIdx.x;
  int a0  = blockIdx.x * 16;

  hf[tid] = ne[(size_t)(NTOT - 1) * HD + tid];
  if (tid < 16) {
    int c = bc[a0 + tid];
    child[tid] = c;
    float t = tv[a0 + tid] * (1.0f / (1.0f + 1e-8f));
    float r = isr[a0 + tid];
    tnl[tid] = t; irl[tid] = r; lgs[tid] = 0.f;
    o_ef[(size_t)(a0 + tid) * EFD + 512] = t;
    o_ef[(size_t)(a0 + tid) * EFD + 513] = r;
  }
  __syncthreads();

  // build edge features: [hf, ht, |hf-ht|, hf*ht]; write to LDS and to d_out
  for (int m = 0; m < 16; ++m) {
    const float* hr = ne + (size_t)child[m] * HD;
    float* er = &ef[m * 518];
    float* og = o_ef + (size_t)(a0 + m) * EFD;
    float a = hf[tid], b = hr[tid];
    float dd = fabsf(a - b), p = a * b;
    er[tid] = a; er[128 + tid] = b; er[256 + tid] = dd; er[384 + tid] = p;
    og[tid] = a; og[128 + tid] = b; og[256 + tid] = dd; og[384 + tid] = p;
  }
  __syncthreads();

  int lane = tid & 31;
  int wave = tid >> 5;
  int n16  = lane & 15;
  int hi   = lane >> 4;
  int koff = hi * 2;

  // z1 = elu(ef @ Wh1 + b1), trailing 2 columns added as rank-1 updates
  for (int nt = wave; nt < 8; nt += 4) {
    int col = nt * 16 + n16;
    v8f acc = {0.f,0.f,0.f,0.f,0.f,0.f,0.f,0.f};
    const float* em = &ef[n16 * 518];
    #pragma unroll 4
    for (int k = 0; k < 512; k += 4) {
      v2f a, b;
      a.x = em[k + koff];
      a.y = em[k + koff + 1];
      const float* wp = Wh1 + (size_t)(k + koff) * HD + col;
      b.x = wp[0];
      b.y = wp[HD];
      acc = __builtin_amdgcn_wmma_f32_16x16x4_f32(false, a, false, b,
                                                  (short)0, acc, false, false);
    }
    float w512 = Wh1[(size_t)512 * HD + col];
    float w513 = Wh1[(size_t)513 * HD + col];
    float bb   = bh1[col];
    #pragma unroll
    for (int r = 0; r < 8; ++r) {
      int m = r + 8 * hi;
      float v = acc[r] + tnl[m] * w512 + irl[m] * w513 + bb;
      z1[m * 132 + col] = elu1(v);
    }
  }
  __syncthreads();

  // z2 = elu(z1 @ Wh2 + b2); logits accumulate z2 * Wh3 via LDS float atomics
  float ll[8];
  #pragma unroll
  for (int r = 0; r < 8; ++r) ll[r] = 0.f;
  for (int nt = wave; nt < 8; nt += 4) {
    int col = nt * 16 + n16;
    v8f acc = {0.f,0.f,0.f,0.f,0.f,0.f,0.f,0.f};
    const float* zm = &z1[n16 * 132];
    #pragma unroll 4
    for (int k = 0; k < 128; k += 4) {
      v2f a, b;
      a.x = zm[k + koff];
      a.y = zm[k + koff + 1];
      const float* wp = Wh2 + (size_t)(k + koff) * HD + col;
      b.x = wp[0];
      b.y = wp[HD];
      acc = __builtin_amdgcn_wmma_f32_16x16x4_f32(false, a, false, b,
                                                  (short)0, acc, false, false);
    }
    float bb = bh2[col];
    float w3 = Wh3[col];
    #pragma unroll
    for (int r = 0; r < 8; ++r) ll[r] += elu1(acc[r] + bb) * w3;
  }
  #pragma unroll
  for (int r = 0; r < 8; ++r) atomicAdd(&lgs[r + 8 * hi], ll[r]);   // ds_add_f32
  __syncthreads();
  if (tid < 16) o_log[a0 + tid] = lgs[tid] + bh3[0];
}

// ---------------------------------------------------------------------------
// softmax over A
// ---------------------------------------------------------------------------
__global__ void redmax_k(const float* __restrict__ l, float* __restrict__ red) {
  __shared__ float sm[256];
  float v = -INFINITY;
  for (int i = blockIdx.x * 256 + threadIdx.x; i < NACT; i += gridDim.x * 256)
    v = fmaxf(v, l[i]);
  sm[threadIdx.x] = v; __syncthreads();
  for (int s = 128; s > 0; s >>= 1) {
    if (threadIdx.x < s) sm[threadIdx.x] = fmaxf(sm[threadIdx.x], sm[threadIdx.x + s]);
    __syncthreads();
  }
  if (threadIdx.x == 0) red[blockIdx.x] = sm[0];
}

__global__ void redmax2_k(float* red) {
  __shared__ float sm[256];
  float v = fmaxf(red[threadIdx.x], red[threadIdx.x + 256]);
  sm[threadIdx.x] = v; __syncthreads();
  for (int s = 128; s > 0; s >>= 1) {
    if (threadIdx.x < s) sm[threadIdx.x] = fmaxf(sm[threadIdx.x], sm[threadIdx.x + s]);
    __syncthreads();
  }
  if (threadIdx.x == 0) { red[512] = sm[0]; red[513] = 0.f; }
}

__global__ void expsum_k(const float* __restrict__ l, float* __restrict__ p,
                         float* __restrict__ red) {
  __shared__ float sm[256];
  int i = blockIdx.x * 256 + threadIdx.x;
  float gmax = red[512];
  float v = 0.f;
  if (i < NACT) { v = expf(l[i] - gmax); p[i] = v; }
  sm[threadIdx.x] = v; __syncthreads();
  for (int s = 128; s > 0; s >>= 1) {
    if (threadIdx.x < s) sm[threadIdx.x] += sm[threadIdx.x + s];
    __syncthreads();
  }
  if (threadIdx.x == 0) atomicAdd(&red[513], sm[0]);
}

__global__ void scale_k(float* __restrict__ p, const float* __restrict__ red) {
  int i = blockIdx.x * 256 + threadIdx.x;
  if (i < NACT) p[i] *= (1.0f / red[513]);
}

__global__ void leaf_k(const int* __restrict__ cfp, float* __restrict__ o) {
  int i = threadIdx.x;
  if (i < LLEAF) o[i] = (i == *cfp) ? 1.f : 0.f;
}

// ---------------------------------------------------------------------------
extern "C" void kernel_launch(void* const* d_in, const int* in_sizes, int n_in,
                              void* d_out, int out_size, void* d_ws, size_t ws_size,
                              hipStream_t stream) {
  (void)in_sizes; (void)n_in; (void)out_size; (void)ws_size;

  const float* x   = (const float*)d_in[0];
  const int*   ei  = (const int*)d_in[1];
  const int*   bc  = (const int*)d_in[2];
  const float* tv  = (const float*)d_in[3];
  const float* isr = (const float*)d_in[4];
  const int*   cf  = (const int*)d_in[5];
  const float* Wg1 = (const float*)d_in[6];
  const float* bg1 = (const float*)d_in[7];
  const float* Wg2 = (const float*)d_in[8];
  const float* bg2 = (const float*)d_in[9];
  const float* Wh1 = (const float*)d_in[10];
  const float* bh1 = (const float*)d_in[11];
  const float* Wh2 = (const float*)d_in[12];
  const float* bh2 = (const float*)d_in[13];
  const float* Wh3 = (const float*)d_in[14];
  const float* bh3 = (const float*)d_in[15];

  float* out    = (float*)d_out;
  float* o_log  = out;
  float* o_prob = out + NACT;
  float* o_ef   = out + 2 * (size_t)NACT;
  float* o_ne   = o_ef + (size_t)NACT * EFD;
  float* o_leaf = o_ne + (size_t)NTOT * HD;

  float* wsf = (float*)d_ws;
  float* isd = wsf;                          // NTPAD floats (deg -> rsqrt in place)
  float* red = wsf + 262144;                 // softmax scratch
  float* m1  = wsf + 524288;                 // NTPAD*128
  float* hb  = m1 + (size_t)NTPAD * HD;      // NTPAD*128

  const size_t gemmLds = 16 * 132 * sizeof(float);   // 8448 B dynamic LDS

  // normalization coefficients
  deg_init_k<<<(NTPAD + 255) / 256, 256, 0, stream>>>(isd);
  deg_accum_k<<<(NEDGE + 255) / 256, 256, 0, stream>>>(ei, isd);
  isd_k<<<(NTPAD + 255) / 256, 256, 0, stream>>>(isd);

  // GCN layer 1
  gemm128_k<<<NNODES / 16, 256, gemmLds, stream>>>(x, Wg1, m1, NNODES / 16);
  focal_k<<<1, 128, 0, stream>>>(Wg1, cf, m1);
  combine1_k<<<(NTPAD * HD) / 256, 256, 0, stream>>>(m1, isd, bg1, hb);
  scatter_k<<<(2 * NEDGE * 32) / 256, 256, 0, stream>>>(ei, isd, m1, hb);
  relu_k<<<(NTPAD * HD) / 256, 256, 0, stream>>>(hb);

  // GCN layer 2 -> node_embeddings directly in d_out
  gemm128_k<<<NTPAD / 16, 256, gemmLds, stream>>>(hb, Wg2, m1, NTPAD / 16);
  combine2_k<<<((size_t)NTOT * HD + 255) / 256, 256, 0, stream>>>(m1, isd, bg2, o_ne);
  scatter_k<<<(2 * NEDGE * 32) / 256, 256, 0, stream>>>(ei, isd, m1, o_ne);

  // fused edge-feature + MLP head
  head_k<<<NACT / 16, 128, 0, stream>>>(o_ne, bc, tv, isr, Wh1, bh1, Wh2, bh2,
                                        Wh3, bh3, o_ef, o_log);

  // softmax
  redmax_k<<<512, 256, 0, stream>>>(o_log, red);
  redmax2_k<<<1, 256, 0, stream>>>(red);
  expsum_k<<<(NACT + 255) / 256, 256, 0, stream>>>(o_log, o_prob, red);
  scale_k<<<(NACT + 255) / 256, 256, 0, stream>>>(o_prob, red);

  // leaf one-hot
  leaf_k<<<1, 128, 0, stream>>>(cf, o_leaf);
}